// Model_54966991454521
// MI455X (gfx1250) — compile-verified
//
#include <hip/hip_runtime.h>

typedef float v2f __attribute__((ext_vector_type(2)));
typedef float v8f __attribute__((ext_vector_type(8)));

#define B_   2
#define C_   64
#define H_   128
#define W_   128
#define NH   8
#define DH   8
#define BW   256        // B_*W_
#define O_   128        // 2*C_
#define REL  255        // 2*H_-1

// workspace layout (float offsets)
#define WS_KQV    0            // 256*128*128 = 4194304
#define WS_SCALE  4194304      // 128
#define WS_SHIFT  4194432      // 128
#define WS_STATS  4194560      // 48  (24 ch x {sum,sumsq})
#define WS_AVEC   4194608      // 24
#define WS_CST    4194632      // 8

// ---------------------------------------------------------------------------
// K1: kqv[bw][o][h] = sum_c kqv_w[o][c] * x[b][c][h][w]   (per bw: 128x128x64)
// f32 WMMA 16x16x4, 8 waves x 8 N-tiles x 16 K-steps.
// ---------------------------------------------------------------------------
__global__ __launch_bounds__(256) void k_kqv_gemm(const float* __restrict__ x,
                                                  const float* __restrict__ kqv_w,
                                                  float* __restrict__ kqv) {
  __shared__ float xs[C_ * H_];                 // 32 KB: x[b][:, :, w]
  const int bw = blockIdx.x;
  const int b0 = bw / W_, w0 = bw % W_;
  const int t = threadIdx.x;
  for (int i = t; i < C_ * H_; i += 256) {
    int c = i >> 7, h = i & 127;
    xs[i] = x[((b0 * C_ + c) * H_ + h) * W_ + w0];
  }
  __syncthreads();

  const int lane = t & 31, wv = t >> 5;
  const int l16 = lane & 15, half = lane >> 4;
  const int M0 = wv * 16;

  v8f acc[8];
#pragma unroll
  for (int n = 0; n < 8; ++n) acc[n] = {};

#pragma unroll
  for (int kk = 0; kk < 16; ++kk) {
    const int k0 = kk * 4;
    // A[M][K]: M = M0 + l16, K = 2*half + j  (j = vgpr index)
    v2f a;
    a.x = kqv_w[(M0 + l16) * C_ + k0 + 2 * half];
    a.y = kqv_w[(M0 + l16) * C_ + k0 + 2 * half + 1];
#pragma unroll
    for (int n = 0; n < 8; ++n) {
      // B[K][N]: K = 2*half + j, N = n*16 + l16
      v2f bf;
      bf.x = xs[(k0 + 2 * half) * H_ + n * 16 + l16];
      bf.y = xs[(k0 + 2 * half + 1) * H_ + n * 16 + l16];
      acc[n] = __builtin_amdgcn_wmma_f32_16x16x4_f32(false, a, false, bf,
                                                     (short)0, acc[n], false, false);
    }
  }
  // D layout: M = M0 + r + 8*half, N = n*16 + l16
#pragma unroll
  for (int n = 0; n < 8; ++n)
#pragma unroll
    for (int r = 0; r < 8; ++r) {
      const int o = M0 + r + 8 * half;
      const int h = n * 16 + l16;
      kqv[(bw * O_ + o) * H_ + h] = acc[n][r];
    }
}

// ---------------------------------------------------------------------------
// K2: per-channel BN over kqv -> folded scale/shift
// ---------------------------------------------------------------------------
__global__ __launch_bounds__(256) void k_kqv_bnstats(const float* __restrict__ kqv,
                                                     const float* __restrict__ g,
                                                     const float* __restrict__ bb,
                                                     float* __restrict__ scale,
                                                     float* __restrict__ shift) {
  __shared__ float rs[256], rss[256];
  const int o = blockIdx.x, t = threadIdx.x;
  float s = 0.f, ss = 0.f;
  for (int i = t; i < BW * H_; i += 256) {
    float v = kqv[(i >> 7) * (O_ * H_) + o * H_ + (i & 127)];
    s += v; ss += v * v;
  }
  rs[t] = s; rss[t] = ss;
  __syncthreads();
  for (int off = 128; off > 0; off >>= 1) {
    if (t < off) { rs[t] += rs[t + off]; rss[t] += rss[t + off]; }
    __syncthreads();
  }
  if (t == 0) {
    const float inv = 1.0f / (float)(BW * H_);
    float mean = rs[0] * inv;
    float var = rss[0] * inv - mean * mean;
    float a = g[o] * rsqrtf(var + 1e-5f);
    scale[o] = a;
    shift[o] = bb[o] - mean * a;
  }
}

// ---------------------------------------------------------------------------
// K3: logits BN statistics: per-24-channel sum/sumsq of qk, qr, kr
// (recomputed scalar; values not stored)
// ---------------------------------------------------------------------------
__global__ __launch_bounds__(256) void k_logits_stats(const float* __restrict__ kqv,
                                                      const float* __restrict__ scale,
                                                      const float* __restrict__ shift,
                                                      const float* __restrict__ re,
                                                      float* __restrict__ stats) {
  __shared__ float ks2[4 * H_], qs[4 * H_], re8[8 * REL];
  const int bw = blockIdx.x >> 3, head = blockIdx.x & 7;
  const int t = threadIdx.x;
  for (int i = t; i < 8 * H_; i += 256) {
    const int cl = i >> 7, h = i & 127;
    const int ch = head * 16 + cl;
    float v = kqv[(bw * O_ + ch) * H_ + h] * scale[ch] + shift[ch];
    if (cl < 4) ks2[cl * H_ + h] = v;            // k: t 0..3
    else        qs[(cl - 4) * H_ + h] = v;       // q: t 4..7
  }
  for (int i = t; i < 8 * REL; i += 256) re8[i] = re[i];
  __syncthreads();

  float sqk = 0, ssqk = 0, sqr = 0, ssqr = 0, skr = 0, sskr = 0;
  for (int xy = t; xy < H_ * H_; xy += 256) {
    const int xx = xy >> 7, yy = xy & 127;
    const int u = yy - xx + 127;
    float qk = 0, qr = 0, kr = 0;
#pragma unroll
    for (int d = 0; d < 4; ++d) {
      qk += qs[d * H_ + xx] * ks2[d * H_ + yy];
      qr += qs[d * H_ + xx] * re8[d * REL + u];
      kr += ks2[d * H_ + yy] * re8[(4 + d) * REL + (254 - u)];
    }
    sqk += qk; ssqk += qk * qk;
    sqr += qr; ssqr += qr * qr;
    skr += kr; sskr += kr * kr;
  }
#pragma unroll
  for (int off = 16; off > 0; off >>= 1) {
    sqk += __shfl_down(sqk, off, 32); ssqk += __shfl_down(ssqk, off, 32);
    sqr += __shfl_down(sqr, off, 32); ssqr += __shfl_down(ssqr, off, 32);
    skr += __shfl_down(skr, off, 32); sskr += __shfl_down(sskr, off, 32);
  }
  if ((t & 31) == 0) {
    atomicAdd(&stats[(0 * NH + head) * 2 + 0], sqk);
    atomicAdd(&stats[(0 * NH + head) * 2 + 1], ssqk);
    atomicAdd(&stats[(1 * NH + head) * 2 + 0], sqr);
    atomicAdd(&stats[(1 * NH + head) * 2 + 1], ssqr);
    atomicAdd(&stats[(2 * NH + head) * 2 + 0], skr);
    atomicAdd(&stats[(2 * NH + head) * 2 + 1], sskr);
  }
}

// ---------------------------------------------------------------------------
// K4: finalize logits BN: a[24] and per-head additive constant
// ---------------------------------------------------------------------------
__global__ __launch_bounds__(32) void k_logits_finalize(const float* __restrict__ stats,
                                                        const float* __restrict__ g,
                                                        const float* __restrict__ bb,
                                                        float* __restrict__ avec,
                                                        float* __restrict__ cst) {
  __shared__ float part[24];
  const int t = threadIdx.x;
  if (t < 24) {
    const float inv = 1.0f / (float)((long long)BW * H_ * H_);
    float mean = stats[t * 2] * inv;
    float var = stats[t * 2 + 1] * inv - mean * mean;
    float a = g[t] * rsqrtf(var + 1e-5f);
    avec[t] = a;
    part[t] = bb[t] - mean * a;
  }
  __syncthreads();
  if (t < 8) cst[t] = part[t] + part[8 + t] + part[16 + t];
}

// ---------------------------------------------------------------------------
// K5: per (bw, head): qk via WMMA, + qr/kr, softmax over y, then
// out[d][x] = sum_y w[x][y] * (v[d][y] + rel[8+d][y-x+127])
// ---------------------------------------------------------------------------
__global__ __launch_bounds__(256) void k_attention(const float* __restrict__ kqv,
                                                   const float* __restrict__ scale,
                                                   const float* __restrict__ shift,
                                                   const float* __restrict__ re,
                                                   const float* __restrict__ avec,
                                                   const float* __restrict__ cst,
                                                   float* __restrict__ out) {
  __shared__ float ks2[4 * H_], qs[4 * H_], vs[8 * H_], rel[16 * REL];
  const int bw = blockIdx.x >> 3, head = blockIdx.x & 7;
  const int b0 = bw / W_, w0 = bw % W_;
  const int t = threadIdx.x;
  for (int i = t; i < 16 * H_; i += 256) {
    const int cl = i >> 7, h = i & 127;
    const int ch = head * 16 + cl;
    float v = kqv[(bw * O_ + ch) * H_ + h] * scale[ch] + shift[ch];
    if (cl < 4)      ks2[cl * H_ + h] = v;
    else if (cl < 8) qs[(cl - 4) * H_ + h] = v;
    else             vs[(cl - 8) * H_ + h] = v;
  }
  for (int i = t; i < 16 * REL; i += 256) rel[i] = re[i];
  __syncthreads();

  const int lane = t & 31, wv = t >> 5;
  const int l16 = lane & 15, half = lane >> 4;
  const int X0 = wv * 16;
  const float aQK = avec[head], aQR = avec[8 + head], aKR = avec[16 + head];
  const float c0 = cst[head];

  // qk: single K=4 WMMA per 16x16 tile. A[x][d] = q[d][x], B[d][y] = k[d][y].
  v2f a;
  a.x = qs[(2 * half) * H_ + X0 + l16];
  a.y = qs[(2 * half + 1) * H_ + X0 + l16];
  v8f acc[8];
#pragma unroll
  for (int n = 0; n < 8; ++n) {
    v2f bf;
    bf.x = ks2[(2 * half) * H_ + n * 16 + l16];
    bf.y = ks2[(2 * half + 1) * H_ + n * 16 + l16];
    v8f cz = {};
    acc[n] = __builtin_amdgcn_wmma_f32_16x16x4_f32(false, a, false, bf,
                                                   (short)0, cz, false, false);
  }

  // logits = aQK*qk + aQR*qr + aKR*kr + c0   (element (n,r): x=X0+r+8*half, y=n*16+l16)
#pragma unroll
  for (int n = 0; n < 8; ++n) {
    const int yy = n * 16 + l16;
#pragma unroll
    for (int r = 0; r < 8; ++r) {
      const int xx = X0 + r + 8 * half;
      const int u = yy - xx + 127;
      float qr = 0, kr = 0;
#pragma unroll
      for (int d = 0; d < 4; ++d) {
        qr += qs[d * H_ + xx] * rel[d * REL + u];
        kr += ks2[d * H_ + yy] * rel[(4 + d) * REL + (254 - u)];
      }
      acc[n][r] = aQK * acc[n][r] + aQR * qr + aKR * kr + c0;
    }
  }

  // softmax over y: each x-row lives in one 16-lane half across the 8 tiles
#pragma unroll
  for (int r = 0; r < 8; ++r) {
    float m = -3.4e38f;
#pragma unroll
    for (int n = 0; n < 8; ++n) m = fmaxf(m, acc[n][r]);
#pragma unroll
    for (int off = 8; off > 0; off >>= 1) m = fmaxf(m, __shfl_xor(m, off, 16));
    float s = 0.f;
#pragma unroll
    for (int n = 0; n < 8; ++n) { float e = __expf(acc[n][r] - m); acc[n][r] = e; s += e; }
#pragma unroll
    for (int off = 8; off > 0; off >>= 1) s += __shfl_xor(s, off, 16);
    const float inv = 1.0f / s;
#pragma unroll
    for (int n = 0; n < 8; ++n) acc[n][r] *= inv;
  }

  // out[d][x] = sum_y w[x][y] * (v[d][y] + v_enc[d][x][y])
#pragma unroll
  for (int d = 0; d < 8; ++d) {
#pragma unroll
    for (int r = 0; r < 8; ++r) {
      const int xx = X0 + r + 8 * half;
      float po = 0.f;
#pragma unroll
      for (int n = 0; n < 8; ++n) {
        const int yy = n * 16 + l16;
        po += acc[n][r] * (vs[d * H_ + yy] + rel[(8 + d) * REL + (yy - xx + 127)]);
      }
#pragma unroll
      for (int off = 8; off > 0; off >>= 1) po += __shfl_xor(po, off, 16);
      if (l16 == 0)
        out[((b0 * C_ + head * DH + d) * H_ + xx) * W_ + w0] = po;
    }
  }
}

// ---------------------------------------------------------------------------
extern "C" void kernel_launch(void* const* d_in, const int* in_sizes, int n_in,
                              void* d_out, int out_size, void* d_ws, size_t ws_size,
                              hipStream_t stream) {
  const float* x     = (const float*)d_in[0];
  const float* kqv_w = (const float*)d_in[1];
  const float* kqv_g = (const float*)d_in[2];
  const float* kqv_b = (const float*)d_in[3];
  const float* log_g = (const float*)d_in[4];
  const float* log_b = (const float*)d_in[5];
  const float* re    = (const float*)d_in[6];
  float* out = (float*)d_out;
  float* ws  = (float*)d_ws;

  float* kqv   = ws + WS_KQV;
  float* scale = ws + WS_SCALE;
  float* shift = ws + WS_SHIFT;
  float* stats = ws + WS_STATS;
  float* avec  = ws + WS_AVEC;
  float* cstv  = ws + WS_CST;

  hipMemsetAsync(stats, 0, 48 * sizeof(float), stream);
  k_kqv_gemm<<<BW, 256, 0, stream>>>(x, kqv_w, kqv);
  k_kqv_bnstats<<<O_, 256, 0, stream>>>(kqv, kqv_g, kqv_b, scale, shift);
  k_logits_stats<<<BW * NH, 256, 0, stream>>>(kqv, scale, shift, re, stats);
  k_logits_finalize<<<1, 32, 0, stream>>>(stats, log_g, log_b, avec, cstv);
  k_attention<<<BW * NH, 256, 0, stream>>>(kqv, scale, shift, re, avec, cstv, out);
}